// CTCLoss_57836029607957
// MI455X (gfx1250) — compile-verified
//
#include <hip/hip_runtime.h>
#include <stdint.h>

// Problem constants (match the reference)
#define CTC_B     256
#define CTC_T     1024
#define CTC_C     128
#define CTC_U     100
#define CTC_S     201          // 2U+1 extended label length
#define CTC_BLANK 127
#define CTC_NEG   (-1.0e30f)
#define CTC_EPS   (1e-7f)

// Async global->LDS pipeline: ring of 8 tiles (512B each), 4 tiles in flight.
#define PIPE_DEPTH 8
#define PIPE_LOOK  4
#define NTHREADS   224         // 7 wave32s; covers S=201 positions

typedef int v4i __attribute__((__vector_size__(4 * sizeof(int))));
typedef __attribute__((address_space(1))) v4i as1_v4i;
typedef __attribute__((address_space(3))) v4i as3_v4i;
typedef __attribute__((address_space(3))) char as3_char;

// One call: each of the 32 lanes of the executing wave copies 16 bytes
// global -> LDS asynchronously (tracked by ASYNCcnt, not LOADcnt).
__device__ __forceinline__ void async_copy16(const float* gsrc, float* ldst) {
#if __has_builtin(__builtin_amdgcn_global_load_async_to_lds_b128)
  __builtin_amdgcn_global_load_async_to_lds_b128(
      (as1_v4i*)(void*)gsrc, (as3_v4i*)ldst, /*offset=*/0, /*cpol=*/0);
#else
  unsigned loff = (unsigned)(__SIZE_TYPE__)(as3_char*)ldst;  // LDS byte offset
  asm volatile("global_load_async_to_lds_b128 %0, %1, off"
               :: "v"(loff), "v"(gsrc) : "memory");
#endif
}

#if __has_builtin(__builtin_amdgcn_s_wait_asynccnt)
#define WAIT_ASYNC_LE(n) __builtin_amdgcn_s_wait_asynccnt(n)
#else
#define WAIT_ASYNC_LE(n) asm volatile("s_wait_asynccnt %0" :: "i"(n))
#endif

__global__ __launch_bounds__(NTHREADS) void CTCLoss_57836029607957_kernel(
    const float* __restrict__ y_pred,   // [B,T,C] post-softmax probs
    const int*   __restrict__ y_true,   // [B,U]
    const int*   __restrict__ label_len,// [B]
    float*       __restrict__ out)      // [B,1]
{
  __shared__ float prob[PIPE_DEPTH][CTC_C];   // async-DMA'd prob rows (ring)
  __shared__ float alpha[2][CTC_S + 7];       // double-buffered DP state

  const int b    = blockIdx.x;
  const int s    = threadIdx.x;
  const int lane = threadIdx.x & 31;
  const bool issuer = (threadIdx.x < 32);     // wave 0 drives the DMA pipeline
  const float* base = y_pred + (size_t)b * CTC_T * CTC_C;

  // --- per-position static data: extended label + skip-transition flag ---
  int  ext  = CTC_BLANK;
  bool skip = false;
  if (s < CTC_S && (s & 1)) {
    ext = y_true[b * CTC_U + (s >> 1)];
    if (s >= 3) skip = (ext != y_true[b * CTC_U + ((s - 3) >> 1)]);
  }

  // --- t = 0 init: only s=0 (blank) and s=1 (first label) reachable ---
  float my_alpha = CTC_NEG;
  if (s < CTC_S) {
    if (s <= 1) my_alpha = __logf(base[ext] + CTC_EPS);
    alpha[0][s] = my_alpha;
  }

  // --- prologue: launch tiles t = 1 .. PIPE_LOOK into the LDS ring ---
  if (issuer) {
    #pragma unroll
    for (int t = 1; t <= PIPE_LOOK; ++t) {
      async_copy16(base + (size_t)t * CTC_C + lane * 4,
                   &prob[t % PIPE_DEPTH][lane * 4]);
    }
  }

  // --- main scan over time: one barrier per step ---
  for (int t = 1; t < CTC_T; ++t) {
    if (issuer) {
      int tf = t + PIPE_LOOK;                   // prefetch target
      if (tf > CTC_T - 1) tf = CTC_T - 1;       // clamp keeps count uniform
      async_copy16(base + (size_t)tf * CTC_C + lane * 4,
                   &prob[(t + PIPE_LOOK) % PIPE_DEPTH][lane * 4]);
      WAIT_ASYNC_LE(PIPE_LOOK);                 // oldest tile (t) has landed
    }
    __syncthreads();  // publishes tile t AND previous alpha buffer

    if (s < CTC_S) {
      const int rb = (t - 1) & 1;
      const float a0 = my_alpha;
      const float a1 = (s >= 1) ? alpha[rb][s - 1] : CTC_NEG;
      const float a2 = skip     ? alpha[rb][s - 2] : CTC_NEG;
      const float m   = fmaxf(a0, fmaxf(a1, a2));
      const float sum = __expf(a0 - m) + __expf(a1 - m) + __expf(a2 - m);
      const float lp  = __logf(prob[t % PIPE_DEPTH][ext] + CTC_EPS);
      my_alpha = m + __logf(sum) + lp;
      alpha[t & 1][s] = my_alpha;
    }
  }

  __syncthreads();
  // --- epilogue: loss = -logsumexp(alpha[2L-1], alpha[2L]) ---
  if (s == 0) {
    const int   end = 2 * label_len[b];         // in [2, 200], S-1 = 200
    const int   lb  = (CTC_T - 1) & 1;          // buffer written at t = T-1
    const float x = alpha[lb][end - 1];
    const float y = alpha[lb][end];
    const float m = fmaxf(x, y);
    out[b] = -(m + __logf(__expf(x - m) + __expf(y - m)));
  }
}

extern "C" void kernel_launch(void* const* d_in, const int* in_sizes, int n_in,
                              void* d_out, int out_size, void* d_ws, size_t ws_size,
                              hipStream_t stream) {
  (void)in_sizes; (void)n_in; (void)out_size; (void)d_ws; (void)ws_size;
  const float* y_pred    = (const float*)d_in[0];
  const int*   y_true    = (const int*)d_in[1];
  const int*   label_len = (const int*)d_in[2];
  float*       out       = (float*)d_out;
  CTCLoss_57836029607957_kernel<<<CTC_B, NTHREADS, 0, stream>>>(
      y_pred, y_true, label_len, out);
}